// GraphMessagePassing_7335804142018
// MI455X (gfx1250) — compile-verified
//
#include <hip/hip_runtime.h>

typedef __attribute__((ext_vector_type(2))) float v2f;
typedef __attribute__((ext_vector_type(4))) float v4f;
typedef __attribute__((ext_vector_type(8))) float v8f;

#define N_NODES 100000
#define N_EDGES 1600000
#define DIM     128

// ---------------------------------------------------------------------------
// GEMM: out[n, j] = relu( sum_k in[n, k] * W[j, k] + bias[j] )  (+ resid[n, j])
// One wave computes one 16x16 (nodes x features) tile via V_WMMA_F32_16X16X4_F32.
// Block = 256 threads = 8 waves -> 8 feature tiles (128 features) per 16-node tile.
// Grid.x = N_NODES/16 = 6250 (exact, no tail -> EXEC all-ones for WMMA).
// NOTE: `in` may alias `out` (GEMM2 runs in place on d_out); rows are
// block-exclusive and the __syncthreads() separates all A-reads from D-stores.
// ---------------------------------------------------------------------------
template <bool RESIDUAL>
__global__ __launch_bounds__(256) void gemm_relu_kernel(
    const float* in, const float* __restrict__ W,
    const float* __restrict__ bias, const float* __restrict__ resid,
    float* out) {
  const int lane = threadIdx.x & 31;
  const int wave = threadIdx.x >> 5;   // 0..7 -> feature tile
  const int row0 = blockIdx.x * 16;    // node tile base
  const int j0   = wave * 16;          // feature tile base
  const int mn   = lane & 15;          // m for A, n for B/C/D
  const int half = lane >> 4;          // K sub-select for A/B, row+8 for C/D

  // A: a[v] = in[row0+m][k0 + 2*half + v]  -> contiguous float2
  // B: b[v] = W[j0+n][k0 + 2*half + v]     -> contiguous float2 (B = W^T tile)
  const float* aptr = in + (size_t)(row0 + mn) * DIM + half * 2;
  const float* bptr = W  + (size_t)(j0 + mn)  * DIM + half * 2;

  v8f c = {};
#pragma unroll
  for (int k0 = 0; k0 < DIM; k0 += 4) {
    v2f a = *(const v2f*)(aptr + k0);
    v2f b = *(const v2f*)(bptr + k0);
    // (neg_a, A, neg_b, B, c_mod, C, reuse_a, reuse_b)
    c = __builtin_amdgcn_wmma_f32_16x16x4_f32(
        false, a, false, b, (short)0, c, false, false);
  }

  // In-place safety: all waves of this block finish reading rows row0..row0+15
  // before any wave stores into those rows.
  __syncthreads();

  const float bj = bias[j0 + mn];
#pragma unroll
  for (int v = 0; v < 8; ++v) {
    const int m = v + half * 8;                       // C/D row for this vgpr
    const size_t idx = (size_t)(row0 + m) * DIM + j0 + mn;
    float val = fmaxf(c[v] + bj, 0.0f);
    if (RESIDUAL) val += resid[idx];
    out[idx] = val;
  }
}

// ---------------------------------------------------------------------------
// Zero the aggregation buffer (float4-wide). 12.8M floats = 3.2M float4.
// ---------------------------------------------------------------------------
__global__ __launch_bounds__(256) void zero_kernel(v4f* __restrict__ p) {
  const size_t i = (size_t)blockIdx.x * blockDim.x + threadIdx.x;
  p[i] = v4f{0.0f, 0.0f, 0.0f, 0.0f};
}

// ---------------------------------------------------------------------------
// Edge scatter: aggregated[dst] += messages[src], one wave per edge.
// Lane l gathers float4 messages[src*128 + 4l .. 4l+3] (coalesced 512B/edge)
// and issues 4 hardware global_atomic_add_f32 into aggregated[dst].
// Both arrays are 51.2 MB -> L2-resident (192 MB L2) -> L2-side atomics.
// ---------------------------------------------------------------------------
__global__ __launch_bounds__(256) void scatter_kernel(
    const float* __restrict__ messages,
    const long long* __restrict__ edge_index,  // [2, N_EDGES] int64
    float* __restrict__ aggregated) {
  const int gid  = blockIdx.x * blockDim.x + threadIdx.x;
  const int edge = gid >> 5;
  const int lane = threadIdx.x & 31;
  if (edge >= N_EDGES) return;

  const long long src = edge_index[edge];
  const long long dst = edge_index[(size_t)N_EDGES + edge];

  v4f m = *(const v4f*)(messages + (size_t)src * DIM + lane * 4);
  float* d = aggregated + (size_t)dst * DIM + lane * 4;
  unsafeAtomicAdd(d + 0, m.x);   // -> global_atomic_add_f32
  unsafeAtomicAdd(d + 1, m.y);
  unsafeAtomicAdd(d + 2, m.z);
  unsafeAtomicAdd(d + 3, m.w);
}

// ---------------------------------------------------------------------------
// Orchestration:
//   messages (d_ws, 51.2 MB)  = relu(x @ W_msg^T + b_msg)        [WMMA]
//   d_out                      = 0
//   d_out[dst]                += messages[src] for all edges      [atomics]
//   d_out                      = relu(d_out @ W_upd^T + b_upd)+x  [WMMA, in place]
// ---------------------------------------------------------------------------
extern "C" void kernel_launch(void* const* d_in, const int* in_sizes, int n_in,
                              void* d_out, int out_size, void* d_ws, size_t ws_size,
                              hipStream_t stream) {
  const float*     x          = (const float*)d_in[0];
  const long long* edge_index = (const long long*)d_in[1];
  const float*     W_msg      = (const float*)d_in[2];
  const float*     b_msg      = (const float*)d_in[3];
  const float*     W_upd      = (const float*)d_in[4];
  const float*     b_upd      = (const float*)d_in[5];

  float* out      = (float*)d_out;
  float* messages = (float*)d_ws;   // N_NODES*DIM floats = 51.2 MB

  const int gemm_blocks = N_NODES / 16;                 // 6250
  const int zero_blocks = (N_NODES * DIM / 4) / 256;    // 12500
  const int scat_blocks = (N_EDGES * 32) / 256;         // 200000

  // GEMM1: messages = relu(x @ W_msg^T + b_msg)
  gemm_relu_kernel<false><<<gemm_blocks, 256, 0, stream>>>(
      x, W_msg, b_msg, nullptr, messages);

  // aggregated (aliased onto d_out) = 0
  zero_kernel<<<zero_blocks, 256, 0, stream>>>((v4f*)out);

  // aggregated[dst] += messages[src]
  scatter_kernel<<<scat_blocks, 256, 0, stream>>>(messages, edge_index, out);

  // out = relu(aggregated @ W_upd^T + b_upd) + x   (in place on d_out)
  gemm_relu_kernel<true><<<gemm_blocks, 256, 0, stream>>>(
      out, W_upd, b_upd, x, out);
}